// AWLoss1D_27719718929043
// MI455X (gfx1250) — compile-verified
//
#include <hip/hip_runtime.h>

typedef __attribute__((ext_vector_type(2))) float v2f;
typedef __attribute__((ext_vector_type(8))) float v8f;

#define Hn   256
#define LDA  260          // 256 + 4 pad: row stride rotates 4 banks -> conflict-light column walks
#define NB   16
#define EPSI 1e-4f

__device__ __forceinline__ float& As(float* A, int i, int j) { return A[i * LDA + j]; }

// One workgroup (256 threads = 8 wave32) per batch. Whole 256x256 system lives in LDS (~273KB < 320KB/WGP).
extern "C" __global__ void __launch_bounds__(256)
awloss_batch(const float* __restrict__ recon, const float* __restrict__ target,
             float* __restrict__ partial)
{
    extern __shared__ float smem[];
    float* A   = smem;                 // 256*260 floats
    float* tg  = smem + Hn * LDA;      // 256
    float* rc  = tg + Hn;              // 256
    float* rr  = rc + Hn;              // 256 autocorrelation
    float* bb  = rr + Hn;              // 256 rhs
    float* ys  = bb + Hn;              // 256 fwd-subst result
    float* vs  = ys + Hn;              // 256 solution
    float* red = vs + Hn;              // 256 reduction scratch

    const int b    = blockIdx.x;
    const int tid  = threadIdx.x;
    const int wave = tid >> 5;
    const int lane = tid & 31;

    tg[tid] = target[b * Hn + tid];
    rc[tid] = recon[b * Hn + tid];
    __syncthreads();

    // ---- autocorrelation r[l] = sum_m tg[m]*tg[m+l]  (DtD is Toeplitz(r)) ----
    {
        float acc = 0.f;
        const int l = tid;
        for (int m = 0; m + l < Hn; ++m) acc += tg[m] * tg[m + l];
        rr[l] = acc;
    }
    // ---- rhs b[j] = sum_m tg[m] * recon[m + j - 127]  (pad_l = 127) ----
    {
        float acc = 0.f;
        const int j   = tid;
        const int mlo = (127 - j) > 0 ? (127 - j) : 0;
        const int mhi = (Hn + 127 - j) < Hn ? (Hn + 127 - j) : Hn;
        for (int m = mlo; m < mhi; ++m) acc += tg[m] * rc[m + j - 127];
        bb[j] = acc;
    }
    __syncthreads();

    // ---- build A = Toeplitz(rr) + eps*I (full symmetric), row per thread ----
    for (int j = 0; j < Hn; ++j) {
        int d = tid - j; d = d < 0 ? -d : d;
        float v = rr[d];
        if (j == tid) v += EPSI;
        As(A, tid, j) = v;
    }
    __syncthreads();

    // ================= blocked right-looking Cholesky, NB = 16 =================
    for (int k0 = 0; k0 < Hn; k0 += NB) {
        // --- (1) factor 16x16 diagonal block: lanes 0..15 of wave 0 (lockstep) ---
        if (tid < NB) {
            const int i = tid;
            for (int j = 0; j < NB; ++j) {
                float djj = sqrtf(As(A, k0 + j, k0 + j));
                if (i >= j) {
                    float lij = As(A, k0 + i, k0 + j) / djj;   // i==j -> sqrt(d)
                    As(A, k0 + i, k0 + j) = lij;
                    for (int c = j + 1; c <= i; ++c)           // row-i rank-1 update
                        As(A, k0 + i, k0 + c) -= lij * As(A, k0 + c, k0 + j);
                }
            }
        }
        __syncthreads();

        // --- (2) panel TRSM: L21 = A21 * L11^{-T}, one row per thread ---
        {
            const int nr = Hn - k0 - NB;
            if (tid < nr) {
                const int i = k0 + NB + tid;
                float p[NB];
                #pragma unroll
                for (int c = 0; c < NB; ++c) p[c] = As(A, i, k0 + c);
                #pragma unroll
                for (int j = 0; j < NB; ++j) {
                    float t = p[j];
                    #pragma unroll
                    for (int m = 0; m < j; ++m) t -= p[m] * As(A, k0 + j, k0 + m);
                    p[j] = t / As(A, k0 + j, k0 + j);
                }
                #pragma unroll
                for (int c = 0; c < NB; ++c) As(A, i, k0 + c) = p[c];
            }
        }
        __syncthreads();

        // --- (3) trailing update A22 -= L21*L21^T : one 16x16 tile per wave, FP32 WMMA ---
        {
            const int tblk   = (Hn - k0 - NB) / NB;
            const int ntiles = tblk * (tblk + 1) / 2;          // lower-triangular tiles
            const int hl     = lane >> 4;                      // half-wave select
            const int l15    = lane & 15;
            const int m0     = hl * 8;                         // C/D: M = vgpr + 8*hl, N = l15
            for (int tile = wave; tile < ntiles; tile += 8) {  // wave-uniform -> EXEC all 1s
                int bi = 0;
                while ((bi + 1) * (bi + 2) / 2 <= tile) ++bi;
                const int bj = tile - bi * (bi + 1) / 2;
                const int i0 = k0 + NB + bi * NB;
                const int j0 = k0 + NB + bj * NB;

                v8f c;
                #pragma unroll
                for (int r = 0; r < 8; ++r) c[r] = As(A, i0 + m0 + r, j0 + l15);

                const int rowA = i0 + l15;                     // A frag: lanes -> M
                const int rowB = j0 + l15;                     // B frag: lanes -> N (B = P_j^T)
                #pragma unroll
                for (int s = 0; s < 4; ++s) {                  // K=16 as 4 x (16x16x4)
                    const int cb = k0 + 4 * s + 2 * hl;        // K split {0,1}/{2,3} per half-wave
                    v2f af, bf;
                    af.x = -As(A, rowA, cb);                   // negate A: f32 WMMA has CNeg only
                    af.y = -As(A, rowA, cb + 1);
                    bf.x =  As(A, rowB, cb);
                    bf.y =  As(A, rowB, cb + 1);
                    c = __builtin_amdgcn_wmma_f32_16x16x4_f32(
                            false, af, false, bf, (short)0, c, false, false);
                }
                #pragma unroll
                for (int r = 0; r < 8; ++r) As(A, i0 + m0 + r, j0 + l15) = c[r];
            }
        }
        __syncthreads();
    }

    // ---- forward substitution L y = b ----
    {
        float acc = 0.f;
        for (int j = 0; j < Hn; ++j) {
            if (tid == j) ys[j] = (bb[j] - acc) / As(A, j, j);
            __syncthreads();
            if (tid > j) acc += As(A, tid, j) * ys[j];
        }
    }
    __syncthreads();
    // ---- back substitution L^T v = y ----
    {
        float acc = 0.f;
        for (int j = Hn - 1; j >= 0; --j) {
            if (tid == j) vs[j] = (ys[j] - acc) / As(A, j, j);
            __syncthreads();
            if (tid < j) acc += As(A, j, tid) * vs[j];
        }
    }
    __syncthreads();

    // ---- T vector (faithful: -gaussian, + max|g|, / max) + norms + loss ----
    const float dx  = 20.0f / 255.0f;
    const float arg = dx * ((float)tid + 0.5f) - 10.0f;
    const float e   = expf(-0.5f * arg * arg);                 // |g| before shifts

    red[tid] = e; __syncthreads();
    for (int s = 128; s > 0; s >>= 1) { if (tid < s) red[tid] = fmaxf(red[tid], red[tid + s]); __syncthreads(); }
    const float m1 = red[0]; __syncthreads();
    const float g  = m1 - e;                                    // g = -exp(..) + max|g|

    red[tid] = g; __syncthreads();
    for (int s = 128; s > 0; s >>= 1) { if (tid < s) red[tid] = fmaxf(red[tid], red[tid + s]); __syncthreads(); }
    const float m2 = red[0]; __syncthreads();
    const float Ti = g / m2;

    const float vv = vs[tid];
    const float tv = Ti * vv;

    red[tid] = tv * tv; __syncthreads();
    for (int s = 128; s > 0; s >>= 1) { if (tid < s) red[tid] += red[tid + s]; __syncthreads(); }
    const float num2 = red[0]; __syncthreads();

    red[tid] = vv * vv; __syncthreads();
    for (int s = 128; s > 0; s >>= 1) { if (tid < s) red[tid] += red[tid + s]; __syncthreads(); }
    const float den2 = red[0];

    if (tid == 0) partial[b] = 0.5f * sqrtf(num2) / sqrtf(den2);
}

// Deterministic tree-sum of per-batch partials (no float atomics).
extern "C" __global__ void __launch_bounds__(256)
awloss_reduce(const float* __restrict__ partial, float* __restrict__ out, int n)
{
    __shared__ float s[256];
    const int tid = threadIdx.x;
    float a = 0.f;
    for (int i = tid; i < n; i += 256) a += partial[i];
    s[tid] = a; __syncthreads();
    for (int k = 128; k > 0; k >>= 1) { if (tid < k) s[tid] += s[tid + k]; __syncthreads(); }
    if (tid == 0) out[0] = s[0];
}

extern "C" void kernel_launch(void* const* d_in, const int* in_sizes, int n_in,
                              void* d_out, int out_size, void* d_ws, size_t ws_size,
                              hipStream_t stream)
{
    const float* recon  = (const float*)d_in[0];
    const float* target = (const float*)d_in[1];
    float* out     = (float*)d_out;
    float* partial = (float*)d_ws;                 // B floats of scratch
    const int B = in_sizes[0] / Hn;                // 512

    const size_t smemBytes = (size_t)(Hn * LDA + 7 * Hn) * sizeof(float); // ~273 KB < 320 KB/WGP

    hipLaunchKernelGGL(awloss_batch, dim3(B), dim3(256), smemBytes, stream,
                       recon, target, partial);
    hipLaunchKernelGGL(awloss_reduce, dim3(1), dim3(256), 0, stream,
                       partial, out, B);
}